// MultiHeadedAttention_54391465836711
// MI455X (gfx1250) — compile-verified
//
#include <hip/hip_runtime.h>
#include <hip/hip_bf16.h>
#include <stdint.h>

// Problem constants (match reference)
constexpr int B_   = 4;
constexpr int D_   = 512;
constexpr int N_   = 1536;
constexpr int H_   = 8;
constexpr int DIM_ = 64;          // per-head dim
constexpr int NT_  = N_ / 16;     // 96 tiles along n/m
constexpr int OT_  = D_ / 16;     // 32 tiles along output channels
constexpr int SP_  = 1540;        // padded LDS row stride (floats), not mult of 64

typedef __attribute__((ext_vector_type(2))) float v2f;
typedef __attribute__((ext_vector_type(8))) float v8f;

static __device__ __forceinline__ v8f wmma_f32(v2f a, v2f b, v8f c) {
  // V_WMMA_F32_16X16X4_F32 : D = A(16x4) * B(4x16) + C(16x16), fp32
  return __builtin_amdgcn_wmma_f32_16x16x4_f32(false, a, false, b, (short)0, c,
                                               false, false);
}

// ---------------------------------------------------------------------------
// Kernel 1: fused Q/K/V projections.
//   out[o,n] = sum_c W[o,c] * x[b,c,n] + bias[o]
// stored head-major: ws[b][h][n][d] with o = d*H + h  (matches reshape(B,DIM,H,N))
// One 16x16 output tile per wave; K loop = 512/4 WMMA steps.
// ---------------------------------------------------------------------------
__global__ void k_proj(const float* __restrict__ xq, const float* __restrict__ xk,
                       const float* __restrict__ xv,
                       const float* __restrict__ Wq, const float* __restrict__ bq,
                       const float* __restrict__ Wk, const float* __restrict__ bk,
                       const float* __restrict__ Wv, const float* __restrict__ bv,
                       float* __restrict__ Qws, float* __restrict__ Kws,
                       float* __restrict__ Vws) {
  const int lane = threadIdx.x & 31;
  const int wave = threadIdx.x >> 5;
  const int lr = lane & 15;   // A: row M ; B: col N ; C: col
  const int lk = lane >> 4;   // k-pair selector

  int t = blockIdx.x * 8 + wave;
  int nt = t % NT_; t /= NT_;
  int ot = t % OT_; t /= OT_;
  int b  = t % B_;  t /= B_;
  int p  = t;                               // 0=q 1=k 2=v

  const float *X, *W, *bias; float* Out;
  if (p == 0)      { X = xq; W = Wq; bias = bq; Out = Qws; }
  else if (p == 1) { X = xk; W = Wk; bias = bk; Out = Kws; }
  else             { X = xv; W = Wv; bias = bv; Out = Vws; }

  const int ob = ot * 16, nb = nt * 16;
  v8f c = {};

  const float* wrow = W + (size_t)(ob + lr) * D_ + 2 * lk;                 // contiguous pair
  const float* xcol = X + (size_t)b * D_ * N_ + (size_t)(2 * lk) * N_ + nb + lr;

  for (int c0 = 0; c0 < D_; c0 += 4) {
    v2f a = *(const v2f*)(wrow + c0);
    v2f bf;
    bf.x = xcol[(size_t)c0 * N_];
    bf.y = xcol[(size_t)(c0 + 1) * N_];
    c = wmma_f32(a, bf, c);
  }

#pragma unroll
  for (int r = 0; r < 8; ++r) {
    const int o = ob + r + 8 * lk;       // C row M = r + 8*(lane/16)
    const int n = nb + lr;               // C col
    const int h = o & 7, d = o >> 3;     // o = d*H + h
    Out[(((size_t)b * H_ + h) * N_ + n) * DIM_ + d] = c[r] + bias[o];
  }
}

// ---------------------------------------------------------------------------
// Kernel 2: global min over scaled scores (no storage).  Block = (b,h,ntile).
// ---------------------------------------------------------------------------
__global__ void k_minpass(const float* __restrict__ Qws, const float* __restrict__ Kws,
                          unsigned* __restrict__ minenc) {
  __shared__ float red[8];
  const int lane = threadIdx.x & 31, wave = threadIdx.x >> 5;
  const int lr = lane & 15, lk = lane >> 4;

  int t = blockIdx.x;
  int nt = t % NT_; t /= NT_;
  int h  = t % H_;  t /= H_;
  int b  = t;

  const float* Qh = Qws + (size_t)(b * H_ + h) * N_ * DIM_;
  const float* Kh = Kws + (size_t)(b * H_ + h) * N_ * DIM_;

  v2f a[16];
  {
    const float* qr = Qh + (size_t)(nt * 16 + lr) * DIM_ + 2 * lk;
#pragma unroll
    for (int kk = 0; kk < 16; ++kk) a[kk] = *(const v2f*)(qr + kk * 4);
  }

  float mn = 3.4e38f;
  for (int mt = wave; mt < NT_; mt += 8) {
    const float* kr = Kh + (size_t)(mt * 16 + lr) * DIM_ + 2 * lk;
    v8f c = {};
#pragma unroll
    for (int kk = 0; kk < 16; ++kk) {
      v2f bf = *(const v2f*)(kr + kk * 4);
      c = wmma_f32(a[kk], bf, c);
    }
#pragma unroll
    for (int r = 0; r < 8; ++r) mn = fminf(mn, c[r] * 0.125f);
  }
  for (int off = 16; off > 0; off >>= 1) mn = fminf(mn, __shfl_xor(mn, off, 32));
  if (lane == 0) red[wave] = mn;
  __syncthreads();
  if (threadIdx.x == 0) {
    float m = red[0];
#pragma unroll
    for (int i = 1; i < 8; ++i) m = fminf(m, red[i]);
    unsigned bits = __float_as_uint(m);
    unsigned msk  = (bits & 0x80000000u) ? 0xFFFFFFFFu : 0x80000000u;
    atomicMin(minenc, bits ^ msk);   // monotonic-uint encoding
  }
}

// ---------------------------------------------------------------------------
// Kernel 3: attention. Block = (b, 16-row n-tile). Loops h=0..7.
//  - recompute 16x1536 score strip into LDS (scaled + penalty)
//  - accumulate head-mean into d_out (owner-computes RMW)
//  - exact row softmax in LDS
//  - P*V via WMMA (4 d-tiles x 2 k-halves, combined through LDS)
//  X stored as [b][n][c] with c = d*H + h.
// ---------------------------------------------------------------------------
__global__ void k_attn(const float* __restrict__ Qws, const float* __restrict__ Kws,
                       const float* __restrict__ Vws,
                       const unsigned char* __restrict__ mask,
                       const unsigned* __restrict__ minenc,
                       float* __restrict__ smean, float* __restrict__ Xws) {
  extern __shared__ float strip[];        // 16 * SP_ floats
  __shared__ float cbuf[4][256];

  const int lane = threadIdx.x & 31, wave = threadIdx.x >> 5;
  const int lr = lane & 15, lk = lane >> 4;
  const int b  = blockIdx.x / NT_;
  const int nb = (blockIdx.x % NT_) * 16;

  unsigned e = *minenc;
  unsigned bits = (e & 0x80000000u) ? (e ^ 0x80000000u) : ~e;
  const float pen = __uint_as_float(bits) - 20.0f;

  for (int h = 0; h < H_; ++h) {
    const float* Qh = Qws + (size_t)(b * H_ + h) * N_ * DIM_;
    const float* Kh = Kws + (size_t)(b * H_ + h) * N_ * DIM_;
    const float* Vh = Vws + (size_t)(b * H_ + h) * N_ * DIM_;

    v2f a[16];
    {
      const float* qr = Qh + (size_t)(nb + lr) * DIM_ + 2 * lk;
#pragma unroll
      for (int kk = 0; kk < 16; ++kk) a[kk] = *(const v2f*)(qr + kk * 4);
    }

    // ---- score strip ----
    for (int mt = wave; mt < NT_; mt += 8) {
      const float* kr = Kh + (size_t)(mt * 16 + lr) * DIM_ + 2 * lk;
      v8f c = {};
#pragma unroll
      for (int kk = 0; kk < 16; ++kk) {
        v2f bf = *(const v2f*)(kr + kk * 4);
        c = wmma_f32(a[kk], bf, c);
      }
      const int m = mt * 16 + lr;
#pragma unroll
      for (int r = 0; r < 8; ++r) {
        const int nl = r + 8 * lk;
        float s = c[r] * 0.125f;
        if (!mask[(size_t)(nb + nl) * N_ + m]) s += pen;
        strip[nl * SP_ + m] = s;
      }
    }
    __syncthreads();

    // ---- accumulate head-mean of penalized scores ----
    {
      float* srow = smean + ((size_t)b * N_ + nb) * N_;
      for (int idx = threadIdx.x; idx < 16 * (N_ / 4); idx += 256) {
        const int row = idx / (N_ / 4);
        const int c4  = (idx % (N_ / 4)) * 4;
        const float4 vv = *(const float4*)(&strip[row * SP_ + c4]);
        float* gp = srow + (size_t)row * N_ + c4;
        float4 res;
        if (h == 0) {
          res.x = vv.x * 0.125f; res.y = vv.y * 0.125f;
          res.z = vv.z * 0.125f; res.w = vv.w * 0.125f;
        } else {
          float4 old = *(const float4*)gp;
          res.x = old.x + vv.x * 0.125f; res.y = old.y + vv.y * 0.125f;
          res.z = old.z + vv.z * 0.125f; res.w = old.w + vv.w * 0.125f;
        }
        *(float4*)gp = res;
      }
    }
    __syncthreads();

    // ---- row softmax (2 rows per wave) ----
#pragma unroll
    for (int rr = 0; rr < 2; ++rr) {
      const int row = wave * 2 + rr;
      float mx = -3.4e38f;
      for (int j = lane; j < N_; j += 32) mx = fmaxf(mx, strip[row * SP_ + j]);
      for (int off = 16; off > 0; off >>= 1) mx = fmaxf(mx, __shfl_xor(mx, off, 32));
      float sum = 0.f;
      for (int j = lane; j < N_; j += 32) {
        float p = expf(strip[row * SP_ + j] - mx);
        strip[row * SP_ + j] = p;
        sum += p;
      }
      for (int off = 16; off > 0; off >>= 1) sum += __shfl_xor(sum, off, 32);
      const float inv = 1.0f / sum;
      for (int j = lane; j < N_; j += 32) strip[row * SP_ + j] *= inv;
    }
    __syncthreads();

    // ---- P @ V : wave -> (d-tile = wave&3, k-half = wave>>2) ----
    {
      const int dt = wave & 3, kh = wave >> 2;
      const int d0 = dt * 16;
      v8f c = {};
      const int m0b = kh * (N_ / 2);
      for (int m0 = m0b; m0 < m0b + N_ / 2; m0 += 4) {
        v2f a2 = *(const v2f*)(&strip[lr * SP_ + m0 + 2 * lk]);   // P rows
        const float* vp = Vh + (size_t)(m0 + 2 * lk) * DIM_ + d0 + lr;
        v2f bf; bf.x = vp[0]; bf.y = vp[DIM_];
        c = wmma_f32(a2, bf, c);
      }
      if (kh == 1) {
#pragma unroll
        for (int r = 0; r < 8; ++r) cbuf[dt][r * 32 + lane] = c[r];
      }
      __syncthreads();
      if (kh == 0) {
#pragma unroll
        for (int r = 0; r < 8; ++r) {
          const float v = c[r] + cbuf[dt][r * 32 + lane];
          const int nl = r + 8 * lk;                       // n row in tile
          Xws[((size_t)b * N_ + nb + nl) * D_ + (size_t)(d0 + lr) * H_ + h] = v;
        }
      }
    }
    __syncthreads();
  }
}

// ---------------------------------------------------------------------------
// Kernel 4: output projection  out[b,o,n] = sum_c Wm[o,c]*Xws[b,n,c] + bm[o]
// ---------------------------------------------------------------------------
__global__ void k_out(const float* __restrict__ Xws, const float* __restrict__ Wm,
                      const float* __restrict__ bm, float* __restrict__ out1) {
  const int lane = threadIdx.x & 31, wave = threadIdx.x >> 5;
  const int lr = lane & 15, lk = lane >> 4;

  int t = blockIdx.x * 8 + wave;
  int nt = t % NT_; t /= NT_;
  int ot = t % OT_; t /= OT_;
  int b  = t;

  const int ob = ot * 16, nb = nt * 16;
  v8f c = {};
  const float* wrow = Wm  + (size_t)(ob + lr) * D_ + 2 * lk;
  const float* xrow = Xws + ((size_t)b * N_ + nb + lr) * D_ + 2 * lk;
  for (int c0 = 0; c0 < D_; c0 += 4) {
    v2f a  = *(const v2f*)(wrow + c0);
    v2f bf = *(const v2f*)(xrow + c0);   // contiguous in c
    c = wmma_f32(a, bf, c);
  }
#pragma unroll
  for (int r = 0; r < 8; ++r) {
    const int o = ob + r + 8 * lk;
    out1[((size_t)b * D_ + o) * N_ + nb + lr] = c[r] + bm[o];
  }
}

// ---------------------------------------------------------------------------
extern "C" void kernel_launch(void* const* d_in, const int* in_sizes, int n_in,
                              void* d_out, int out_size, void* d_ws, size_t ws_size,
                              hipStream_t stream) {
  const float* query = (const float*)d_in[0];
  const float* key   = (const float*)d_in[1];
  const float* value = (const float*)d_in[2];
  // d_in[3] = dist  (unused, as in reference)
  const unsigned char* mask = (const unsigned char*)d_in[4];
  const float* Wq = (const float*)d_in[5];
  const float* bq = (const float*)d_in[6];
  const float* Wk = (const float*)d_in[7];
  const float* bk = (const float*)d_in[8];
  const float* Wv = (const float*)d_in[9];
  const float* bv = (const float*)d_in[10];
  const float* Wm = (const float*)d_in[11];
  const float* bm = (const float*)d_in[12];

  float* out1  = (float*)d_out;                       // [B, D, N]
  float* smean = out1 + (size_t)B_ * D_ * N_;         // [B, N, N]

  const size_t qkv = (size_t)B_ * H_ * N_ * DIM_;     // == B*D*N
  float* Qws = (float*)d_ws;
  float* Kws = Qws + qkv;
  float* Vws = Kws + qkv;
  float* Xws = Vws + qkv;                             // [B, N, D]
  unsigned* minenc = (unsigned*)(Xws + (size_t)B_ * N_ * D_);

  hipMemsetAsync(minenc, 0xFF, sizeof(unsigned), stream);  // encoded +inf

  // 3 projections * B * 32 * 96 tiles / 8 waves per block
  k_proj<<<3 * B_ * OT_ * NT_ / 8, 256, 0, stream>>>(query, key, value,
                                                     Wq, bq, Wk, bk, Wv, bv,
                                                     Qws, Kws, Vws);
  k_minpass<<<B_ * H_ * NT_, 256, 0, stream>>>(Qws, Kws, minenc);
  k_attn<<<B_ * NT_, 256, 16 * SP_ * sizeof(float), stream>>>(
      Qws, Kws, Vws, mask, minenc, smean, Xws);
  k_out<<<B_ * OT_ * NT_ / 8, 256, 0, stream>>>(Xws, Wm, bm, out1);
}